// HunyuanMoE_46394236731644
// MI455X (gfx1250) — compile-verified
//
#include <hip/hip_runtime.h>

typedef __attribute__((ext_vector_type(2))) float v2f;
typedef __attribute__((ext_vector_type(8))) float v8f;

#define TT 8192   // tokens
#define HH 1024   // hidden
#define II 512    // intermediate
#define EE 16     // experts

__device__ __forceinline__ v8f wmma_k4(v2f a, v2f b, v8f c) {
  // D = A(16x4 f32) * B(4x16 f32) + C(16x16 f32), full fp32 precision
  return __builtin_amdgcn_wmma_f32_16x16x4_f32(false, a, false, b, (short)0, c,
                                               false, false);
}

__device__ __forceinline__ v8f vzero8() {
  v8f z = {0.f, 0.f, 0.f, 0.f, 0.f, 0.f, 0.f, 0.f};
  return z;
}

// ---------------------------------------------------------------------------
// Stage 1: router — logits, softmax(fp32), top-2, renormalize
// ---------------------------------------------------------------------------
__global__ __launch_bounds__(256) void router_kernel(
    const float* __restrict__ x, const float* __restrict__ gate_w,
    int* __restrict__ sel, float* __restrict__ rw) {
  int t = blockIdx.x * blockDim.x + threadIdx.x;
  if (t >= TT) return;
  float acc[EE];
#pragma unroll
  for (int e = 0; e < EE; ++e) acc[e] = 0.f;
  const float* xr = x + (size_t)t * HH;
  for (int h = 0; h < HH; h += 4) {
    float4 xv = *(const float4*)(xr + h);
#pragma unroll
    for (int e = 0; e < EE; ++e) {
      const float* g = gate_w + (size_t)e * HH + h;  // uniform across lanes
      acc[e] += xv.x * g[0] + xv.y * g[1] + xv.z * g[2] + xv.w * g[3];
    }
  }
  // softmax: denominators cancel under top-2 renormalization
  float m = acc[0];
#pragma unroll
  for (int e = 1; e < EE; ++e) m = fmaxf(m, acc[e]);
  float p[EE];
#pragma unroll
  for (int e = 0; e < EE; ++e) p[e] = __expf(acc[e] - m);
  int i0 = 0;
  float p0 = p[0];
#pragma unroll
  for (int e = 1; e < EE; ++e)
    if (p[e] > p0) { p0 = p[e]; i0 = e; }   // strict >: lowest index on ties
  int i1 = -1;
  float p1 = -1.f;
#pragma unroll
  for (int e = 0; e < EE; ++e)
    if (e != i0 && p[e] > p1) { p1 = p[e]; i1 = e; }
  float inv = 1.f / (p0 + p1);
  sel[t * 2 + 0] = i0;
  sel[t * 2 + 1] = i1;
  rw[t * 2 + 0] = p0 * inv;
  rw[t * 2 + 1] = p1 * inv;
}

// ---------------------------------------------------------------------------
// Stage 2: deterministic per-(expert,slot) token list compaction (wave32)
// ---------------------------------------------------------------------------
__global__ void build_lists_kernel(const int* __restrict__ sel,
                                   int* __restrict__ lists,
                                   int* __restrict__ counts) {
  int slab = blockIdx.x;  // e*2 + k
  int e = slab >> 1, k = slab & 1;
  int lane = threadIdx.x;  // 0..31
  int base = 0;
  for (int c = 0; c < TT; c += 32) {
    int t = c + lane;
    bool active = (sel[t * 2 + k] == e);
    unsigned mm = (unsigned)__ballot(active);
    int rank = __popc(mm & ((1u << lane) - 1u));
    if (active) lists[(size_t)slab * TT + base + rank] = t;
    base += __popc(mm);
  }
  if (lane == 0) counts[slab] = base;
}

// ---------------------------------------------------------------------------
// Stage 3: gate_up + SiLU*up (*routing weight).  moe==0: shared expert.
// 16-token tile per workgroup; 8 waves x (64 gate + 64 up) columns.
// ---------------------------------------------------------------------------
__global__ __launch_bounds__(256) void gateup_kernel(
    const float* __restrict__ x, const float* __restrict__ wgu_all,
    const float* __restrict__ bias_all, const int* __restrict__ lists,
    const int* __restrict__ counts, const float* __restrict__ rw,
    float* __restrict__ actbuf, int moe) {
  const int TILES = TT / 16;
  int slab, tile;
  const float *wgu, *bias;
  if (moe) {
    slab = blockIdx.x / TILES;
    tile = blockIdx.x - slab * TILES;
    if (tile * 16 >= counts[slab]) return;  // uniform exit: EXEC stays all-1s
    wgu = wgu_all + (size_t)(slab >> 1) * (2 * II) * HH;
    bias = bias_all + (size_t)(slab >> 1) * (2 * II);
  } else {
    slab = -1;
    tile = blockIdx.x;
    wgu = wgu_all;
    bias = bias_all;
  }

  __shared__ __attribute__((aligned(16))) float ldsX[16 * 516];  // pad->no conflicts
  __shared__ int ldsTok[16];
  __shared__ float ldsScale[16];
  __shared__ int ldsRow[16];

  int tid = threadIdx.x;
  if (tid < 16) {
    int tok = -1, rowid = -1;
    float sc = 1.f;
    if (moe) {
      int idx = tile * 16 + tid;
      if (idx < counts[slab]) {
        tok = lists[(size_t)slab * TT + idx];
        int k = slab & 1;
        sc = rw[tok * 2 + k];
        rowid = tok * 2 + k;
      }
    } else {
      tok = tile * 16 + tid;
      rowid = tok;
    }
    ldsTok[tid] = tok;
    ldsScale[tid] = sc;
    ldsRow[tid] = rowid;
  }
  __syncthreads();

  int wave = tid >> 5, lane = tid & 31;
  int r16 = lane & 15;
  int khalf = (lane >> 4) << 1;  // 0 or 2 (WMMA f32 A/B K split)

  v8f accG[4], accU[4];
#pragma unroll
  for (int s = 0; s < 4; ++s) { accG[s] = vzero8(); accU[s] = vzero8(); }

  for (int kc = 0; kc < HH; kc += 512) {
    // stage gathered x chunk [16 x 512] into LDS, coalesced float4
    for (int idx = tid; idx < 16 * 128; idx += 256) {
      int r = idx >> 7, c4 = idx & 127;
      int tok = ldsTok[r];
      float4 v = make_float4(0.f, 0.f, 0.f, 0.f);
      if (tok >= 0) v = ((const float4*)(x + (size_t)tok * HH + kc))[c4];
      float* dst = &ldsX[r * 516 + c4 * 4];
      dst[0] = v.x; dst[1] = v.y; dst[2] = v.z; dst[3] = v.w;
    }
    __syncthreads();

    const float* arow = &ldsX[r16 * 516];
    for (int kb = 0; kb < 512; kb += 4) {
      int kk = kb + khalf;
      v2f a = *(const v2f*)(arow + kk);
#pragma unroll
      for (int s = 0; s < 4; ++s) {
        int cg = wave * 64 + s * 16 + r16;  // gate column
        v2f bg = *(const v2f*)(wgu + (size_t)cg * HH + kc + kk);
        accG[s] = wmma_k4(a, bg, accG[s]);
        v2f bu = *(const v2f*)(wgu + (size_t)(II + cg) * HH + kc + kk);
        accU[s] = wmma_k4(a, bu, accU[s]);
      }
    }
    __syncthreads();
  }

  // epilogue: bias, SiLU(gate)*up*scale, scatter rows to act buffer
#pragma unroll
  for (int s = 0; s < 4; ++s) {
    int cg = wave * 64 + s * 16 + r16;
    float bG = bias[cg];
    float bU = bias[II + cg];
#pragma unroll
    for (int v = 0; v < 8; ++v) {
      int row = v + ((lane >> 4) << 3);
      int rowid = ldsRow[row];
      if (rowid < 0) continue;
      float g = accG[s][v] + bG;
      float u = accU[s][v] + bU;
      float a = (g / (1.f + __expf(-g))) * u * ldsScale[row];
      actbuf[(size_t)rowid * II + cg] = a;
    }
  }
}

// ---------------------------------------------------------------------------
// Stage 4: down projection.  moe==0: overwrite out (shared expert pass).
// moe==1: out += contribution for slot kslot (each token row touched once).
// ---------------------------------------------------------------------------
__global__ __launch_bounds__(256) void down_kernel(
    const float* __restrict__ wd_all, const int* __restrict__ lists,
    const int* __restrict__ counts, const float* __restrict__ actbuf,
    float* __restrict__ out, int moe, int kslot) {
  const int TILES = TT / 16;
  int tile, slab;
  const float* wd;
  if (moe) {
    int eidx = blockIdx.x / TILES;
    tile = blockIdx.x - eidx * TILES;
    slab = eidx * 2 + kslot;
    if (tile * 16 >= counts[slab]) return;
    wd = wd_all + (size_t)eidx * HH * II;
  } else {
    slab = -1;
    tile = blockIdx.x;
    wd = wd_all;
  }

  __shared__ __attribute__((aligned(16))) float ldsA[16 * 516];
  __shared__ int ldsTok[16];
  __shared__ int ldsRow[16];

  int tid = threadIdx.x;
  if (tid < 16) {
    int tok = -1, rowid = -1;
    if (moe) {
      int idx = tile * 16 + tid;
      if (idx < counts[slab]) {
        tok = lists[(size_t)slab * TT + idx];
        rowid = tok * 2 + kslot;
      }
    } else {
      tok = tile * 16 + tid;
      rowid = tok;
    }
    ldsTok[tid] = tok;
    ldsRow[tid] = rowid;
  }
  __syncthreads();

  // stage act tile [16 x 512] into LDS
  for (int idx = tid; idx < 16 * 128; idx += 256) {
    int r = idx >> 7, c4 = idx & 127;
    int rowid = ldsRow[r];
    float4 v = make_float4(0.f, 0.f, 0.f, 0.f);
    if (rowid >= 0) v = ((const float4*)(actbuf + (size_t)rowid * II))[c4];
    float* dst = &ldsA[r * 516 + c4 * 4];
    dst[0] = v.x; dst[1] = v.y; dst[2] = v.z; dst[3] = v.w;
  }
  __syncthreads();

  int wave = tid >> 5, lane = tid & 31;
  int r16 = lane & 15;
  int khalf = (lane >> 4) << 1;

  v8f acc[8];
#pragma unroll
  for (int s = 0; s < 8; ++s) acc[s] = vzero8();

  const float* arow = &ldsA[r16 * 516];
  for (int kb = 0; kb < II; kb += 4) {
    int kk = kb + khalf;
    v2f a = *(const v2f*)(arow + kk);
#pragma unroll
    for (int s = 0; s < 8; ++s) {
      int ch = wave * 128 + s * 16 + r16;  // output column (H)
      v2f b = *(const v2f*)(wd + (size_t)ch * II + kk);
      acc[s] = wmma_k4(a, b, acc[s]);
    }
  }

#pragma unroll
  for (int s = 0; s < 8; ++s) {
    int ch = wave * 128 + s * 16 + r16;
#pragma unroll
    for (int v = 0; v < 8; ++v) {
      int row = v + ((lane >> 4) << 3);
      int tok = ldsTok[row];
      if (tok < 0) continue;
      size_t o = (size_t)tok * HH + ch;
      if (moe)
        out[o] += acc[s][v];  // race-free within a pass; passes stream-ordered
      else
        out[o] = acc[s][v];   // shared pass initializes every output element
    }
  }
}

// ---------------------------------------------------------------------------
extern "C" void kernel_launch(void* const* d_in, const int* in_sizes, int n_in,
                              void* d_out, int out_size, void* d_ws,
                              size_t ws_size, hipStream_t stream) {
  (void)in_sizes; (void)n_in; (void)out_size; (void)ws_size;
  const float* x         = (const float*)d_in[0];  // [T,H]
  const float* gate_w    = (const float*)d_in[1];  // [E,H]
  const float* w_gate_up = (const float*)d_in[2];  // [E,2I,H]
  const float* b_gate_up = (const float*)d_in[3];  // [E,2I]
  const float* w_down    = (const float*)d_in[4];  // [E,H,I]
  const float* ws_gu     = (const float*)d_in[5];  // [2I,H]
  const float* bs_gu     = (const float*)d_in[6];  // [2I]
  const float* ws_dn     = (const float*)d_in[7];  // [H,I]
  float* out = (float*)d_out;

  // workspace layout (bytes):
  //   [0, 32MB)          act buffer: rows = t*2+k (MoE) / t (shared, aliased,
  //                      safe: shared down completes before MoE gateup writes)
  //   [32MB, +64KB)      sel  int[T*2]
  //   [.. , +64KB)       rw   float[T*2]
  //   [.. , +1MB)        lists int[32*T]
  //   [.. , +128B)       counts int[32]
  char* ws = (char*)d_ws;
  float* actbuf = (float*)ws;
  int*   sel    = (int*)(ws + 33554432);
  float* rw     = (float*)(ws + 33554432 + 65536);
  int*   lists  = (int*)(ws + 33554432 + 131072);
  int*   counts = (int*)(ws + 33554432 + 131072 + 1048576);

  dim3 b256(256);
  router_kernel<<<TT / 256, b256, 0, stream>>>(x, gate_w, sel, rw);
  build_lists_kernel<<<EE * 2, dim3(32), 0, stream>>>(sel, lists, counts);

  // shared expert: writes out (=)
  gateup_kernel<<<TT / 16, b256, 0, stream>>>(x, ws_gu, bs_gu, nullptr, nullptr,
                                              nullptr, actbuf, 0);
  down_kernel<<<TT / 16, b256, 0, stream>>>(ws_dn, nullptr, nullptr, actbuf,
                                            out, 0, 0);

  // MoE top-2 (grouped GEMMs; worst-case grids, device-side early exit)
  gateup_kernel<<<(EE * 2) * (TT / 16), b256, 0, stream>>>(
      x, w_gate_up, b_gate_up, lists, counts, rw, actbuf, 1);
  down_kernel<<<EE * (TT / 16), b256, 0, stream>>>(w_down, lists, counts,
                                                   actbuf, out, 1, 0);
  down_kernel<<<EE * (TT / 16), b256, 0, stream>>>(w_down, lists, counts,
                                                   actbuf, out, 1, 1);
}